// EnergyLoss_88149908783213
// MI455X (gfx1250) — compile-verified
//
#include <hip/hip_runtime.h>

// CDNA5 / gfx1250 implementation.
//
// Math restructure:
//   H(n) = 0.5*M - 0.5*sum_d x_{n,d}*S_d + 0.5*(sum_d x_{n,d}^2)*I
// with sample-independent M = sum_d A_d A_d^H, S_d = A_d + A_d^H.
// => per-sample H build is a GEMM  X[N x 8] * S2[8 x 512]  done with
//    V_WMMA_F32_16X16X4_F32 (full f32 precision), one 16-sample tile per block.
// Smallest eigenvector via power iteration on (trace(H)+1)*I - H.
//
// ws layout (floats): [0 .. 4095]   S2[k][c], k<8, c = (i*16+j)*2 + reim
//                     [4096..4607]  Msum[c]

typedef __attribute__((ext_vector_type(2))) float v2f;
typedef __attribute__((ext_vector_type(8))) float v8f;

#define NITER 200

__global__ __launch_bounds__(256) void EnergyLoss_precompute(
    const float* __restrict__ Ar, const float* __restrict__ Ai,
    float* __restrict__ ws, float* __restrict__ out)
{
    const int t = threadIdx.x;
    if (t == 0) out[0] = 0.0f;

    // Msum[i][j] = sum_d sum_k A[d][i][k] * conj(A[d][j][k])   (one entry/thread)
    {
        const int i = t >> 4, j = t & 15;
        float re = 0.f, im = 0.f;
        for (int d = 0; d < 8; ++d) {
            const float* ar = Ar + d * 256;
            const float* ai = Ai + d * 256;
            for (int k = 0; k < 16; ++k) {
                const float a = ar[i*16 + k], b = ai[i*16 + k];
                const float c = ar[j*16 + k], e = ai[j*16 + k];
                re += a * c + b * e;          // Re((a+bi)(c-ei))
                im += b * c - a * e;          // Im((a+bi)(c-ei))
            }
        }
        ws[4096 + (i*16 + j)*2 + 0] = re;
        ws[4096 + (i*16 + j)*2 + 1] = im;
    }

    // S[d][i][j] = A[d][i][j] + conj(A[d][j][i])  (Hermitian), 2048 complex entries
    for (int e = t; e < 2048; e += 256) {
        const int d = e >> 8, rem = e & 255;
        const int i = rem >> 4, j = rem & 15;
        const float re = Ar[d*256 + i*16 + j] + Ar[d*256 + j*16 + i];
        const float im = Ai[d*256 + i*16 + j] - Ai[d*256 + j*16 + i];
        ws[d*512 + (i*16 + j)*2 + 0] = re;
        ws[d*512 + (i*16 + j)*2 + 1] = im;
    }
}

__global__ __launch_bounds__(256) void EnergyLoss_main(
    const float* __restrict__ Ar, const float* __restrict__ Ai,
    const float* __restrict__ X,  const float* __restrict__ ws,
    float* __restrict__ out, int N)
{
    __shared__ float Hflat[16 * 512];   // 32 KB: [local sample m][c]

    const int tid  = threadIdx.x;
    const int wave = tid >> 5;
    const int lane = tid & 31;
    const int l16  = lane & 15;
    const int hi   = (lane >= 16) ? 1 : 0;

    // ---------------- Phase 1: WMMA f32 GEMM  Hflat[m,c] = sum_k X[s,k]*S2[k,c]
    // A operand (16x4 f32): lanes 0-15 carry K = kbase+{0,1}; lanes 16-31 K = kbase+{2,3}.
    {
        int sA = blockIdx.x * 16 + l16;
        if (sA >= N) sA = 0;                       // clamp, keep EXEC all-ones
        const int kA = hi ? 2 : 0;
        v2f a0, a1;
        a0.x = X[sA*8 + kA + 0];  a0.y = X[sA*8 + kA + 1];
        a1.x = X[sA*8 + 4 + kA];  a1.y = X[sA*8 + 5 + kA];

        for (int t = 0; t < 4; ++t) {
            const int ntile = wave * 4 + t;        // 8 waves x 4 tiles = 32 col tiles
            const int col   = ntile * 16 + l16;
            const int kB    = hi ? 2 : 0;
            v2f b0, b1;                            // B (4x16): mirror of A layout
            b0.x = ws[(kB + 0) * 512 + col];
            b0.y = ws[(kB + 1) * 512 + col];
            b1.x = ws[(4 + kB + 0) * 512 + col];
            b1.y = ws[(4 + kB + 1) * 512 + col];

            v8f acc = {};
            acc = __builtin_amdgcn_wmma_f32_16x16x4_f32(
                      false, a0, false, b0, (short)0, acc, false, false);
            acc = __builtin_amdgcn_wmma_f32_16x16x4_f32(
                      false, a1, false, b1, (short)0, acc, false, false);

            // D layout: VGPR r -> (M=r, N=lane) lanes 0-15 ; (M=r+8, N=lane-16) lanes 16-31
            #pragma unroll
            for (int r = 0; r < 8; ++r) {
                const int mo = r + (hi ? 8 : 0);
                Hflat[mo * 512 + col] = acc[r];
            }
        }
    }
    __syncthreads();

    // ---------------- Phase 2: eigensolve. Wave handles 2 samples (one per 16-lane half).
    const int m      = wave * 2 + hi;              // local sample 0..15
    const int sample = blockIdx.x * 16 + m;
    const int sc     = (sample < N) ? sample : 0;
    const int i      = l16;                        // this lane owns row i of H

    float xloc[8], sumx2 = 0.f;
    #pragma unroll
    for (int d = 0; d < 8; ++d) { xloc[d] = X[sc*8 + d]; sumx2 += xloc[d]*xloc[d]; }

    // H row i:  0.5*Msum - 0.5*Hflat + 0.5*sumx2 * I
    float hrr[16], hri[16];
    #pragma unroll
    for (int j = 0; j < 16; ++j) {
        const int c = (i*16 + j) * 2;
        float re = 0.5f * (ws[4096 + c]     - Hflat[m*512 + c]);
        float im = 0.5f * (ws[4096 + c + 1] - Hflat[m*512 + c + 1]);
        if (j == i) re += 0.5f * sumx2;
        hrr[j] = re; hri[j] = im;
    }

    // sigma = trace(H) + 1  (>= lambda_max for PSD H): butterfly over the 16-lane half
    float tr = hrr[i];
    for (int s = 8; s >= 1; s >>= 1) tr += __shfl_xor(tr, s, 32);
    const float sigma = tr + 1.0f;

    // power iteration on B = sigma*I - H  (dominant eigvec == smallest eigvec of H)
    float vr[16], vi[16];
    {
        float n2 = 0.f;
        #pragma unroll
        for (int j = 0; j < 16; ++j) {
            vr[j] = 1.0f + 0.07f * j; vi[j] = 0.02f * (float)(j & 3);
            n2 += vr[j]*vr[j] + vi[j]*vi[j];
        }
        const float inv = rsqrtf(n2);
        #pragma unroll
        for (int j = 0; j < 16; ++j) { vr[j] *= inv; vi[j] *= inv; }
    }

    const int base = lane & 16;                    // lane-group base for shuffles
    for (int it = 0; it < NITER; ++it) {
        float hvr = 0.f, hvi = 0.f;
        #pragma unroll
        for (int j = 0; j < 16; ++j) {             // (H v)_i, complex dot
            hvr += hrr[j]*vr[j] - hri[j]*vi[j];
            hvi += hrr[j]*vi[j] + hri[j]*vr[j];
        }
        const float yr = sigma * vr[i] - hvr;
        const float yi = sigma * vi[i] - hvi;

        float n2 = 0.f;                            // all-gather y -> v, accumulate norm
        #pragma unroll
        for (int j = 0; j < 16; ++j) {
            const float r = __shfl(yr, base + j, 32);
            const float q = __shfl(yi, base + j, 32);
            vr[j] = r; vi[j] = q;
            n2 += r*r + q*q;
        }
        const float inv = rsqrtf(n2);
        #pragma unroll
        for (int j = 0; j < 16; ++j) { vr[j] *= inv; vi[j] *= inv; }
    }

    // ---------------- Phase 3: loss terms.  psi = v (every lane holds full psi).
    float loss = 0.f;
    for (int d = 0; d < 8; ++d) {
        const float* ar = Ar + d*256 + i*16;
        const float* ai = Ai + d*256 + i*16;
        float apr = 0.f, api = 0.f;                // (A_d psi)_i
        #pragma unroll
        for (int j = 0; j < 16; ++j) {
            apr += ar[j]*vr[j] - ai[j]*vi[j];
            api += ar[j]*vi[j] + ai[j]*vr[j];
        }
        float pr = vr[i]*apr + vi[i]*api;          // Re(conj(psi_i) * (A psi)_i)
        float ns = apr*apr + api*api;              // |(A psi)_i|^2
        for (int s = 8; s >= 1; s >>= 1) {
            pr += __shfl_xor(pr, s, 32);
            ns += __shfl_xor(ns, s, 32);
        }
        const float diff = pr - xloc[d];
        loss += diff*diff + 0.1f * (ns - pr*pr);
    }

    if (i == 0 && sample < N) {
        atomicAdd(out, loss / (float)N);
    }
}

extern "C" void kernel_launch(void* const* d_in, const int* in_sizes, int n_in,
                              void* d_out, int out_size, void* d_ws, size_t ws_size,
                              hipStream_t stream) {
    const float* Ar = (const float*)d_in[0];   // A_real  (8,16,16)
    const float* Ai = (const float*)d_in[1];   // A_imag  (8,16,16)
    const float* X  = (const float*)d_in[2];   // X       (N,8)
    float* out = (float*)d_out;
    float* ws  = (float*)d_ws;
    const int N = in_sizes[2] / 8;

    EnergyLoss_precompute<<<1, 256, 0, stream>>>(Ar, Ai, ws, out);
    const int nblocks = (N + 15) / 16;
    EnergyLoss_main<<<nblocks, 256, 0, stream>>>(Ar, Ai, X, ws, out, N);
}